// RefractiveSurface_4252017623120
// MI455X (gfx1250) — compile-verified
//
#include <hip/hip_runtime.h>

// ---------------------------------------------------------------------------
// RefractiveSurface: ray/circular-arc intersection + Snell refraction.
// Memory-bound streaming kernel (32 B/ray, ~40 FLOP/ray -> AI ~1.25 FLOP/B;
// HBM floor ~11.5 us at 23.3 TB/s for 268 MB of traffic).
// Strategy: float4 (2 rays) per lane => global_load_b128 / global_store_b128,
// non-temporal hints on the streams, gfx1250 global_prefetch ahead.
// ---------------------------------------------------------------------------

typedef float f32x4 __attribute__((ext_vector_type(4)));

#define RCURV 50.0f           // surface radius of curvature
#define ETA   (1.0f / 1.5f)   // N1 / N2

// Per-ray math, fp32, branch-free selects (lowers to v_cndmask_b32).
__device__ __forceinline__ void trace_ray(float ox, float oy, float dx, float dy,
                                          float cx,
                                          float& px, float& py,
                                          float& qx, float& qy) {
    // oc = origin - center ; center = (target + R, 0)
    const float ocx = ox - cx;
    const float ocy = oy;

    // quadratic: t^2 + 2 b t + c = 0
    const float b    = fmaf(dx, ocx, dy * ocy);
    const float c    = fmaf(ocx, ocx, fmaf(ocy, ocy, -RCURV * RCURV));
    const float disc = fmaxf(fmaf(b, b, -c), 0.0f);
    const float t    = -b - __builtin_sqrtf(disc);   // near (front) intersection

    // collision point
    px = fmaf(t, dx, ox);
    py = fmaf(t, dy, oy);

    // outward normal, flipped to oppose incoming ray
    float nx = (px - cx) / RCURV;
    float ny = py / RCURV;
    const float d0   = fmaf(nx, dx, ny * dy);
    const bool flip  = d0 > 0.0f;
    nx = flip ? -nx : nx;
    ny = flip ? -ny : ny;

    // Snell refraction, TIR -> reflection
    const float cosi  = -fmaf(nx, dx, ny * dy);
    const float sint2 = (ETA * ETA) * (1.0f - cosi * cosi);
    const float cost  = __builtin_sqrtf(fmaxf(1.0f - sint2, 0.0f));

    const float k  = fmaf(ETA, cosi, -cost);
    const float rx = fmaf(k, nx, ETA * dx);          // refracted
    const float ry = fmaf(k, ny, ETA * dy);

    const float twoci = 2.0f * cosi;
    const float sx = fmaf(twoci, nx, dx);            // reflected (TIR path)
    const float sy = fmaf(twoci, ny, dy);

    const bool tir = sint2 > 1.0f;
    qx = tir ? sx : rx;
    qy = tir ? sy : ry;
}

// Main kernel: one float4 (= 2 rays) per lane. B128 in, B128 out, NT hints.
__global__ __launch_bounds__(256) void refract_pairs_kernel(
    const f32x4* __restrict__ org,     // (N,2) origins, viewed as N/2 float4
    const f32x4* __restrict__ dir,     // (N,2) unit directions, as N/2 float4
    const float* __restrict__ targ,    // scalar target (surface vertex x)
    f32x4* __restrict__ out_cp,        // collision points, N/2 float4
    f32x4* __restrict__ out_rf,        // refracted dirs,  N/2 float4
    int npairs) {
    const int i = blockIdx.x * blockDim.x + threadIdx.x;
    if (i >= npairs) return;

    // uniform scalar: center x = target + R  (s_load_b32 path)
    const float cx = *targ + RCURV;

    // gfx1250 streaming prefetch ~8KB ahead on both input streams
    // (speculative: translation failures at the tail are silently dropped)
    __builtin_prefetch((const char*)(org + i) + 8192, 0, 0);
    __builtin_prefetch((const char*)(dir + i) + 8192, 0, 0);

    const f32x4 o = __builtin_nontemporal_load(org + i);   // global_load_b128 th:NT
    const f32x4 d = __builtin_nontemporal_load(dir + i);

    // Locals (cannot bind refs to ext_vector elements); assembled below.
    float p0x, p0y, q0x, q0y;
    float p1x, p1y, q1x, q1y;
    trace_ray(o.x, o.y, d.x, d.y, cx, p0x, p0y, q0x, q0y);
    trace_ray(o.z, o.w, d.z, d.w, cx, p1x, p1y, q1x, q1y);

    f32x4 cp, rf;
    cp.x = p0x; cp.y = p0y; cp.z = p1x; cp.w = p1y;
    rf.x = q0x; rf.y = q0y; rf.z = q1x; rf.w = q1y;

    __builtin_nontemporal_store(cp, out_cp + i);           // global_store_b128 th:NT
    __builtin_nontemporal_store(rf, out_rf + i);
}

// Scalar tail (only launched if the ray count is odd; N=2^23, so normally not).
__global__ void refract_tail_kernel(const float* __restrict__ org,
                                    const float* __restrict__ dir,
                                    const float* __restrict__ targ,
                                    float* __restrict__ out,
                                    int nrays, int start) {
    const int i = start + blockIdx.x * blockDim.x + threadIdx.x;
    if (i >= nrays) return;
    const float cx = *targ + RCURV;
    float px, py, qx, qy;
    trace_ray(org[2 * i], org[2 * i + 1], dir[2 * i], dir[2 * i + 1], cx,
              px, py, qx, qy);
    out[2 * i]     = px;
    out[2 * i + 1] = py;
    const size_t base = (size_t)2 * (size_t)nrays;
    out[base + 2 * i]     = qx;
    out[base + 2 * i + 1] = qy;
}

extern "C" void kernel_launch(void* const* d_in, const int* in_sizes, int n_in,
                              void* d_out, int out_size, void* d_ws, size_t ws_size,
                              hipStream_t stream) {
    // Inputs (setup_inputs order): rays_origins (N,2) f32, rays_vectors (N,2) f32,
    // target scalar f32. Output: [collision_points (N,2) | refracted (N,2)] f32.
    const float* org_f = (const float*)d_in[0];
    const float* dir_f = (const float*)d_in[1];
    const float* targ  = (const float*)d_in[2];
    float*       out   = (float*)d_out;

    const int nrays  = in_sizes[0] / 2;   // flat elements / 2 components
    const int npairs = nrays / 2;         // 2 rays per float4 lane

    if (npairs > 0) {
        const f32x4* org = (const f32x4*)org_f;
        const f32x4* dir = (const f32x4*)dir_f;
        f32x4* out_cp = (f32x4*)out;
        f32x4* out_rf = (f32x4*)(out + (size_t)2 * (size_t)nrays);

        const int block = 256;                       // 8 wave32 per workgroup
        const int grid  = (npairs + block - 1) / block;
        refract_pairs_kernel<<<grid, block, 0, stream>>>(org, dir, targ,
                                                         out_cp, out_rf, npairs);
    }

    if (nrays & 1) {   // defensive tail; not taken for N = 8,388,608
        refract_tail_kernel<<<1, 32, 0, stream>>>(org_f, dir_f, targ, out,
                                                  nrays, nrays - 1);
    }
}